// OrderBookFlowSolver_64484638982590
// MI455X (gfx1250) — compile-verified
//
#include <hip/hip_runtime.h>
#include <stdint.h>

// ---------------------------------------------------------------------------
// 1-D Lax-Friedrichs Euler solver, time-tiled with REGISTER-resident state.
// Each thread owns 16 contiguous elements in VGPRs; per step it exchanges only
// its edge (rho, v, P) triples with neighbor threads via a parity-buffered LDS
// mailbox (12 dwords of LDS traffic + 1 barrier per step, vs ~160 + 2 before).
// CDNA5 paths: global_load_async_to_lds_b128 / global_store_async_from_lds_b128
// (ASYNCcnt-tracked), s_wait_asynccnt, v_log_f32/v_exp_f32 transcendentals.
// ---------------------------------------------------------------------------

#define N_LEVELS (1 << 20)
#define MASKN    (N_LEVELS - 1)
#define TILE     4096
#define HALO     128
#define INTERIOR (TILE - 2 * HALO)                       /* 3840 */
#define NT       256
#define EPT      (TILE / NT)                             /* 16 elements/thread */
#define NBLOCKS  ((N_LEVELS + INTERIOR - 1) / INTERIOR)  /* 274  */

// Low 32 bits of a generic pointer to a __shared__ object are the LDS byte
// address (LDS aperture lives entirely in the high 32 bits on gfx1250).
__device__ __forceinline__ uint32_t lds_off(const void* p) {
  return (uint32_t)(uintptr_t)p;
}

__device__ __forceinline__ void async_load_b128(uint32_t lds, uint64_t gaddr) {
  asm volatile("global_load_async_to_lds_b128 %0, %1, off"
               :: "v"(lds), "v"(gaddr) : "memory");
}
__device__ __forceinline__ void async_store_b128(uint64_t gaddr, uint32_t lds) {
  asm volatile("global_store_async_from_lds_b128 %0, %1, off"
               :: "v"(gaddr), "v"(lds) : "memory");
}
__device__ __forceinline__ void wait_async0() {
  asm volatile("s_wait_asynccnt 0x0" ::: "memory");
}

__device__ __forceinline__ float pow_gamma(float x) {
  // max(x,0)^1.4 = exp2(1.4*log2(x)); v_log_f32(0) = -inf -> exp2(-inf) = 0.
  return __builtin_amdgcn_exp2f(1.4f * __builtin_amdgcn_logf(fmaxf(x, 0.0f)));
}

__global__ __launch_bounds__(NT)
void OrderBookFlowSolver_64484638982590_kernel(const float* __restrict__ rho0,
                                               const float* __restrict__ v0,
                                               const int* __restrict__ nsp,
                                               float* __restrict__ out) {
  // Staging tile (reused for load and store phases) + edge-exchange mailbox.
  __shared__ float stage[2 * TILE];       // [0,TILE): rho   [TILE,2*TILE): v
  __shared__ float edge[2][6][NT];        // [parity][Lr,Lv,Lp,Rr,Rv,Rp][tid]

  float* SR = stage;
  float* SV = stage + TILE;

  const int tid   = threadIdx.x;
  const int start = (int)blockIdx.x * INTERIOR - HALO;   // may be negative; wraps

  int nsteps = *nsp;                 // uniform scalar load (graph-capture safe)
  if (nsteps > HALO) nsteps = HALO;  // halo supports up to 128 steps
  if (nsteps < 0)    nsteps = 0;

  // ---- async gather of tile (+halo) into LDS; 16B chunks never straddle wrap
#pragma unroll
  for (int j = 0; j < TILE / 4 / NT; ++j) {              // 4 chunks / thread
    int c = tid + NT * j;
    int g = (start + 4 * c) & MASKN;                     // periodic wrap
    async_load_b128(lds_off(&SR[4 * c]), (uint64_t)(uintptr_t)(rho0 + g));
    async_load_b128(lds_off(&SV[4 * c]), (uint64_t)(uintptr_t)(v0 + g));
  }
  wait_async0();
  __syncthreads();

  // ---- pull this thread's 16 contiguous elements into registers
  float r[EPT], v[EPT], p[EPT];
#pragma unroll
  for (int q = 0; q < EPT / 4; ++q) {
    float4 a = ((const float4*)SR)[4 * tid + q];
    float4 b = ((const float4*)SV)[4 * tid + q];
    r[4 * q + 0] = a.x; r[4 * q + 1] = a.y; r[4 * q + 2] = a.z; r[4 * q + 3] = a.w;
    v[4 * q + 0] = b.x; v[4 * q + 1] = b.y; v[4 * q + 2] = b.z; v[4 * q + 3] = b.w;
  }

  const float C1 = 5.0e-3f;   // DT / (2*DX)
  const float CV = 1.0e-2f;   // DT * MU / DX^2

  for (int s = 0; s < nsteps; ++s) {
    const int pb = s & 1;
    // P = max(rho,0)^1.4 for own elements (registers)
#pragma unroll
    for (int k = 0; k < EPT; ++k) p[k] = pow_gamma(r[k]);

    // publish OLD edge triples for neighbors (parity-buffered -> 1 barrier)
    edge[pb][0][tid] = r[0];        edge[pb][3][tid] = r[EPT - 1];
    edge[pb][1][tid] = v[0];        edge[pb][4][tid] = v[EPT - 1];
    edge[pb][2][tid] = p[0];        edge[pb][5][tid] = p[EPT - 1];
    __syncthreads();

    // neighbor halos (tile-edge threads clamp to own edge; garbage there
    // propagates 1 cell/step and is absorbed by the 128-element halo)
    float hlr, hlv, hlp, hrr, hrv, hrp;
    if (tid > 0) {
      hlr = edge[pb][3][tid - 1]; hlv = edge[pb][4][tid - 1]; hlp = edge[pb][5][tid - 1];
    } else { hlr = r[0]; hlv = v[0]; hlp = p[0]; }
    if (tid < NT - 1) {
      hrr = edge[pb][0][tid + 1]; hrv = edge[pb][1][tid + 1]; hrp = edge[pb][2][tid + 1];
    } else { hrr = r[EPT - 1]; hrv = v[EPT - 1]; hrp = p[EPT - 1]; }

    // in-place Lax-Friedrichs sweep over the 16 register elements
    float rm = hlr, vm = hlv, pm = hlp;
#pragma unroll
    for (int k = 0; k < EPT; ++k) {
      float rc = r[k], vc = v[k], pc = p[k];
      float rp2 = (k < EPT - 1) ? r[k + 1] : hrr;
      float vp2 = (k < EPT - 1) ? v[k + 1] : hrv;
      float pp2 = (k < EPT - 1) ? p[k + 1] : hrp;
      float fmm = rm * vm;                         // mass flux @ i-1
      float fmp = rp2 * vp2;                       // mass flux @ i+1
      float rn  = 0.5f * (rp2 + rm) - C1 * (fmp - fmm);
      float fqm = fmm * vm + pm;                   // momentum flux @ i-1
      float fqp = fmp * vp2 + pp2;                 // momentum flux @ i+1
      float mn  = 0.5f * (fmp + fmm) - C1 * (fqp - fqm)
                + CV * (vp2 - 2.0f * vc + vm) * rc;
      rm = rc; vm = vc; pm = pc;                   // save OLD values for next k
      r[k] = rn;
      v[k] = mn / fmaxf(rn, 1e-10f);
    }
    // no second barrier needed: next step writes the other parity slot, and
    // the write-back to this slot (step s+2) is fenced by barrier(s+1).
  }

  // ---- registers -> LDS staging, then async scatter of interior to global
#pragma unroll
  for (int q = 0; q < EPT / 4; ++q) {
    float4 a, b;
    a.x = r[4 * q + 0]; a.y = r[4 * q + 1]; a.z = r[4 * q + 2]; a.w = r[4 * q + 3];
    b.x = v[4 * q + 0]; b.y = v[4 * q + 1]; b.z = v[4 * q + 2]; b.w = v[4 * q + 3];
    ((float4*)SR)[4 * tid + q] = a;
    ((float4*)SV)[4 * tid + q] = b;
  }
  __syncthreads();

#pragma unroll
  for (int j = 0; j < 4; ++j) {
    int c = tid + NT * j;                         // vec4 chunk within tile
    if (c < INTERIOR / 4) {
      int i = HALO + 4 * c;
      int g = start + i;                          // == blockIdx.x*INTERIOR + 4c
      if (g < N_LEVELS) {                         // last block is partial
        async_store_b128((uint64_t)(uintptr_t)(out + g),            lds_off(&SR[i]));
        async_store_b128((uint64_t)(uintptr_t)(out + N_LEVELS + g), lds_off(&SV[i]));
      }
    }
  }
  // S_ENDPGM performs an implicit wait-idle, draining ASYNCcnt.
}

extern "C" void kernel_launch(void* const* d_in, const int* in_sizes, int n_in,
                              void* d_out, int out_size, void* d_ws, size_t ws_size,
                              hipStream_t stream) {
  const float* rho0 = (const float*)d_in[0];
  const float* v0   = (const float*)d_in[1];
  const int*   nsp  = (const int*)d_in[2];
  float* out = (float*)d_out;
  (void)in_sizes; (void)n_in; (void)out_size; (void)d_ws; (void)ws_size;

  OrderBookFlowSolver_64484638982590_kernel<<<NBLOCKS, NT, 0, stream>>>(
      rho0, v0, nsp, out);
}